// HeteroGNN_88098369176018
// MI455X (gfx1250) — compile-verified
//
#include <hip/hip_runtime.h>

// HeteroGNN on MI455X (gfx1250, wave32).
// layer1 = scalar scatter; layer2 = project-then-scatter:
//   segment_sum(h[src]) @ W == segment_sum((h @ W)[src])
// The layer-1 node update (rank-1 expand + relu, since IN==1) is fused INTO the
// WMMA GEMM: the A fragment is synthesized in registers, so the 256MB hidden
// matrices are never written to memory. Dense math uses v_wmma_f32_16x16x4_f32.

#define HIDD 64
#define OUTD 32

typedef float v2f __attribute__((ext_vector_type(2)));
typedef float v8f __attribute__((ext_vector_type(8)));

__device__ __forceinline__ v8f wmma_f32_16x16x4(v2f a, v2f b, v8f c) {
  // D = A(16x4 f32) * B(4x16 f32) + C(16x16 f32)
  return __builtin_amdgcn_wmma_f32_16x16x4_f32(false, a, false, b, (short)0, c,
                                               false, false);
}

// ---- tiny prep: pre-sum the two root weight/bias sets feeding node-type 'a' ----
__global__ __launch_bounds__(256) void prep_kernel(
    const float* __restrict__ Wroot2, const float* __restrict__ b2,
    const float* __restrict__ Wroot1, const float* __restrict__ b1,
    float* __restrict__ buf) {
  int t = blockIdx.x * blockDim.x + threadIdx.x;
  if (t < 2048)       buf[t] = Wroot2[2048 + t] + Wroot2[4096 + t];                // Wroot2[1]+Wroot2[2]
  else if (t < 2080)  buf[t] = b2[32 + (t - 2048)] + b2[64 + (t - 2048)];          // b2[1]+b2[2]
  else if (t < 2144)  buf[t] = Wroot1[64 + (t - 2080)] + Wroot1[128 + (t - 2080)]; // Wroot1[1]+[2]
  else if (t < 2208)  buf[t] = b1[64 + (t - 2144)] + b1[128 + (t - 2144)];         // b1[1]+b1[2]
}

// ---- layer1: scalar scatter-add over edges ----
__global__ __launch_bounds__(256) void l1_scatter(
    const float* __restrict__ x, const int* __restrict__ src,
    const int* __restrict__ dst, float* __restrict__ agg, int E) {
  int e = blockIdx.x * blockDim.x + threadIdx.x;
  if (e >= E) return;
  unsafeAtomicAdd(&agg[dst[e]], x[src[e]]);
}

// ---- fused layer-1 node update + layer-2 dense part.
// One wave owns a 16-row strip. A fragment h[r][c] is computed in registers in
// the exact WMMA A layout (lane m = row, K = kk + 2*(lane/16) + {0,1}), then
// multiplied against NREL rel matrices (-> projections) and the root matrix
// (-> output init, + bias2). NREL==2 <=> node type 'a' (also has 2 layer-1 aggs).
template <int NREL>
__global__ __launch_bounds__(256) void l1_gemm_fused(
    const float* __restrict__ agg0, const float* __restrict__ agg1,
    const float* __restrict__ x,
    const float* __restrict__ wrel0, const float* __restrict__ wrel1,
    const float* __restrict__ wroot1, const float* __restrict__ bias1,
    const float* __restrict__ WrelA, const float* __restrict__ WrelB,
    const float* __restrict__ Wroot2, const float* __restrict__ bias2,
    float* __restrict__ PA, float* __restrict__ PB, float* __restrict__ Oroot,
    int nstrips) {
  const int wave = (int)((blockIdx.x * blockDim.x + threadIdx.x) >> 5);
  if (wave >= nstrips) return;
  const int lane = (int)(threadIdx.x & 31);
  const int m    = lane & 15;            // row within strip (A), col within tile (B/C)
  const int kh   = (lane >> 4) << 1;     // K half: 0 or 2
  const int row0 = wave << 4;
  const int r    = row0 + m;

  // layer-1 per-row scalars (IN == 1)
  const float s0 = agg0[r];
  const float sx = x[r];
  const float s1 = (NREL > 1) ? agg1[r] : 0.0f;

  v8f accR0 = {}, accR1 = {};
  v8f accA0 = {}, accA1 = {};
  v8f accB0 = {}, accB1 = {};

#pragma unroll
  for (int kk = 0; kk < HIDD; kk += 4) {
    const int c = kk + kh;  // even -> 8B-aligned v2f loads

    // synthesize A fragment: h[r][c], h[r][c+1] with relu
    v2f w0 = *(const v2f*)(wrel0 + c);
    v2f wr = *(const v2f*)(wroot1 + c);
    v2f bb = *(const v2f*)(bias1 + c);
    v2f a;
    a.x = s0 * w0.x + sx * wr.x + bb.x;
    a.y = s0 * w0.y + sx * wr.y + bb.y;
    if (NREL > 1) {
      v2f w1 = *(const v2f*)(wrel1 + c);
      a.x += s1 * w1.x;
      a.y += s1 * w1.y;
    }
    a.x = fmaxf(a.x, 0.0f);
    a.y = fmaxf(a.y, 0.0f);

    const int wofs = c * OUTD + m;
    {
      const float* wp = Wroot2 + wofs;
      v2f b0; b0.x = wp[0];  b0.y = wp[OUTD];
      v2f b1; b1.x = wp[16]; b1.y = wp[16 + OUTD];
      accR0 = wmma_f32_16x16x4(a, b0, accR0);
      accR1 = wmma_f32_16x16x4(a, b1, accR1);
    }
    {
      const float* wp = WrelA + wofs;
      v2f b0; b0.x = wp[0];  b0.y = wp[OUTD];
      v2f b1; b1.x = wp[16]; b1.y = wp[16 + OUTD];
      accA0 = wmma_f32_16x16x4(a, b0, accA0);
      accA1 = wmma_f32_16x16x4(a, b1, accA1);
    }
    if (NREL > 1) {
      const float* wp = WrelB + wofs;
      v2f b0; b0.x = wp[0];  b0.y = wp[OUTD];
      v2f b1; b1.x = wp[16]; b1.y = wp[16 + OUTD];
      accB0 = wmma_f32_16x16x4(a, b0, accB0);
      accB1 = wmma_f32_16x16x4(a, b1, accB1);
    }
  }

  // C/D layout: VGPR v -> row = row0 + v + 8*(lane/16), col = (lane&15) [+16]
  const int rbase = row0 + ((lane >> 4) << 3);
#pragma unroll
  for (int v = 0; v < 8; ++v) {
    const size_t ro = (size_t)(rbase + v) * OUTD;
    Oroot[ro + m]      = accR0[v] + bias2[m];
    Oroot[ro + m + 16] = accR1[v] + bias2[m + 16];
    PA[ro + m]         = accA0[v];
    PA[ro + m + 16]    = accA1[v];
    if (NREL > 1) {
      PB[ro + m]      = accB0[v];
      PB[ro + m + 16] = accB1[v];
    }
  }
}

// ---- layer2 scatter: wave per edge, lane j moves one of 32 floats ----
__global__ __launch_bounds__(256) void edge2_scatter(
    const float* __restrict__ P, const int* __restrict__ src,
    const int* __restrict__ dst, float* __restrict__ out, int E) {
  unsigned t = blockIdx.x * blockDim.x + threadIdx.x;
  int e = (int)(t >> 5);
  if (e >= E) return;
  int j = (int)(t & 31);
  int s = src[e], d = dst[e];
  unsafeAtomicAdd(&out[(size_t)d * OUTD + j], P[(size_t)s * OUTD + j]);
}

extern "C" void kernel_launch(void* const* d_in, const int* in_sizes, int n_in,
                              void* d_out, int out_size, void* d_ws, size_t ws_size,
                              hipStream_t stream) {
  const float* x_a    = (const float*)d_in[0];
  const float* x_b    = (const float*)d_in[1];
  const int*   ei_ab  = (const int*)d_in[2];
  const int*   ei_ba  = (const int*)d_in[3];
  const int*   ei_aa  = (const int*)d_in[4];
  const float* Wrel1  = (const float*)d_in[5];
  const float* Wroot1 = (const float*)d_in[6];
  const float* b1     = (const float*)d_in[7];
  const float* Wrel2  = (const float*)d_in[8];
  const float* Wroot2 = (const float*)d_in[9];
  const float* b2     = (const float*)d_in[10];

  const int NA = in_sizes[0];       // x_a is [NA,1]
  const int NB = in_sizes[1];
  const int E  = in_sizes[2] / 2;   // ei is [2,E]

  float* oa = (float*)d_out;                       // [NA,32]
  float* ob = (float*)d_out + (size_t)NA * OUTD;   // [NB,32]

  // workspace layout (floats): prep | aggBA | aggAA | aggAB | P0 | P2 | P1
  float* wsf = (float*)d_ws;
  size_t off = 0;
  float* prep_buf = wsf;               off += 2560;
  float* aggBA = wsf + off;            off += (size_t)NA;
  float* aggAA = wsf + off;            off += (size_t)NA;
  float* aggAB = wsf + off;            off += (size_t)NB;
  float* P0    = wsf + off;            off += (size_t)NA * OUTD;
  float* P2    = wsf + off;            off += (size_t)NA * OUTD;
  float* P1    = wsf + off;            off += (size_t)NB * OUTD;

  // zero the scalar layer-1 aggregation buffers (contiguous)
  hipMemsetAsync(aggBA, 0, (size_t)(2 * NA + NB) * sizeof(float), stream);

  prep_kernel<<<9, 256, 0, stream>>>(Wroot2, b2, Wroot1, b1, prep_buf);

  const float* wroot2_sum = prep_buf;          // 2048
  const float* b2_sum     = prep_buf + 2048;   // 32
  const float* wroot1_sum = prep_buf + 2080;   // 64
  const float* b1_sum     = prep_buf + 2144;   // 64

  // layer-1 scalar scatter (edge rows: [0]=src, [1]=dst)
  int eb = (E + 255) / 256;
  l1_scatter<<<eb, 256, 0, stream>>>(x_b, ei_ba, ei_ba + E, aggBA, E);
  l1_scatter<<<eb, 256, 0, stream>>>(x_a, ei_aa, ei_aa + E, aggAA, E);
  l1_scatter<<<eb, 256, 0, stream>>>(x_a, ei_ab, ei_ab + E, aggAB, E);

  // fused layer-1 node update + layer-2 projections/root (WMMA). N % 16 == 0.
  int strips_a = NA / 16, strips_b = NB / 16;
  // node type 'b': h_b from aggAB/Wrel1[0], Wroot1[0], b1[0];
  //                P1 = h_b @ Wrel2[1]; ob = h_b @ Wroot2[0] + b2[0]
  l1_gemm_fused<1><<<(strips_b + 7) / 8, 256, 0, stream>>>(
      aggAB, nullptr, x_b, Wrel1, nullptr, Wroot1, b1,
      Wrel2 + 1 * HIDD * OUTD, nullptr, Wroot2, b2,
      P1, nullptr, ob, strips_b);
  // node type 'a': h_a from aggBA/Wrel1[1] + aggAA/Wrel1[2], summed roots;
  //                P0 = h_a @ Wrel2[0]; P2 = h_a @ Wrel2[2];
  //                oa = h_a @ (Wroot2[1]+Wroot2[2]) + (b2[1]+b2[2])
  l1_gemm_fused<2><<<(strips_a + 7) / 8, 256, 0, stream>>>(
      aggBA, aggAA, x_a, Wrel1 + HIDD, Wrel1 + 2 * HIDD, wroot1_sum, b1_sum,
      Wrel2 + 0 * HIDD * OUTD, Wrel2 + 2 * HIDD * OUTD, wroot2_sum, b2_sum,
      P0, P2, oa, strips_a);

  // layer-2 scatter of projected rows into outputs
  unsigned e2t = (unsigned)E * 32;
  unsigned e2b = (e2t + 255) / 256;
  edge2_scatter<<<e2b, 256, 0, stream>>>(P0, ei_ab, ei_ab + E, ob, E);  // a->b
  edge2_scatter<<<e2b, 256, 0, stream>>>(P1, ei_ba, ei_ba + E, oa, E);  // b->a
  edge2_scatter<<<e2b, 256, 0, stream>>>(P2, ei_aa, ei_aa + E, oa, E);  // a->a
}